// XentBoeck_37546604102464
// MI455X (gfx1250) — compile-verified
//
#include <hip/hip_runtime.h>
#include <hip/hip_bf16.h>

typedef float f4v __attribute__((ext_vector_type(4)));
typedef float v2f __attribute__((ext_vector_type(2)));
typedef float v8f __attribute__((ext_vector_type(8)));

#define L2E 1.4426950408889634f
#define LN2 0.6931471805599453f
#define NT 512  // threads per block (16 waves of 32)

__device__ __forceinline__ float e2(float x) { return __builtin_amdgcn_exp2f(x); }

// One block per row: online softmax over C elements, then windowed-gather loss.
__global__ __launch_bounds__(NT) void xent_boeck_row(
    const float* __restrict__ preds, const int* __restrict__ labels,
    const float* __restrict__ win, int Lwin, float* __restrict__ row_loss, int C) {
  const int row  = blockIdx.x;
  const int tid  = threadIdx.x;
  const int lane = tid & 31;
  const int wave = tid >> 5;

  const float* base = preds + (long long)row * C;

  // Row base is only 4B-aligned in general (C*4 = 201028 B); peel to 16B.
  const uintptr_t addr = (uintptr_t)base;
  const int head = (int)(((16u - (addr & 15u)) & 15u) >> 2);   // 0..3 floats
  const int n4   = (C - head) >> 2;
  const int tbase = head + (n4 << 2);

  float m = -INFINITY, s = 0.0f;

  // head scalars
  if (tid < head) {
    float x = base[tid];
    float nm = fmaxf(m, x);
    s = s * e2((m - nm) * L2E) + e2((x - nm) * L2E);
    m = nm;
  }

  // main vectorized stream (non-temporal: single-touch data, skip L2 residency)
  const f4v* __restrict__ p = (const f4v*)(base + head);
  for (int i = tid; i < n4; i += NT) {
    f4v v = __builtin_nontemporal_load(p + i);
    float cm = fmaxf(fmaxf(v.x, v.y), fmaxf(v.z, v.w));
    if (cm > m) {            // lazy rescale: rare after warm-up -> ~1 exp/element
      s *= e2((m - cm) * L2E);
      m = cm;
    }
    s += e2((v.x - m) * L2E) + e2((v.y - m) * L2E) +
         e2((v.z - m) * L2E) + e2((v.w - m) * L2E);
  }

  // tail scalars
  if (tid < C - tbase) {
    float x = base[tbase + tid];
    float nm = fmaxf(m, x);
    s = s * e2((m - nm) * L2E) + e2((x - nm) * L2E);
    m = nm;
  }

  __shared__ float smax[16];
  __shared__ float ssum[16];

  // 1) block max: wave32 shuffle reduce, then combine 16 wave maxes
  float wm = m;
  #pragma unroll
  for (int o = 16; o > 0; o >>= 1) wm = fmaxf(wm, __shfl_xor(wm, o, 32));
  if (lane == 0) smax[wave] = wm;
  __syncthreads();
  float M = smax[0];
  #pragma unroll
  for (int i = 1; i < 16; ++i) M = fmaxf(M, smax[i]);

  // 2) rescale each thread's partial sum to the common max
  float sp = s * e2((m - M) * L2E);

  // 3) wave-level sum of 32 partials with ONE v_wmma_f32_16x16x4_f32:
  //    B == ones => D[r,n] = sum_k A[r,k]; every A slot contributes exactly once,
  //    so total over rows == total over the wave's 64 A slots (32 sp's + 32 zeros).
  //    C/D layout: lanes 0-15 hold rows 0-7 in d0..d7, lanes 16-31 hold rows 8-15.
  v2f a; a.x = sp;   a.y = 0.0f;
  v2f b; b.x = 1.0f; b.y = 1.0f;
  v8f c = {};
  c = __builtin_amdgcn_wmma_f32_16x16x4_f32(false, a, false, b, (short)0, c,
                                            false, false);
  float t = c[0] + c[1] + c[2] + c[3] + c[4] + c[5] + c[6] + c[7];
  float wave_sum = t + __shfl_xor(t, 16, 32);   // rows 0-7 + rows 8-15

  if (lane == 0) ssum[wave] = wave_sum;
  __syncthreads();

  // 4) thread 0: combine 16 wave sums (fixed order -> deterministic), finish row
  if (tid == 0) {
    float S = 0.0f;
    #pragma unroll
    for (int i = 0; i < 16; ++i) S += ssum[i];
    float lse = M + LN2 * __builtin_amdgcn_logf(S);

    int   idx  = labels[row];
    int   half = Lwin >> 1;
    float dot = 0.0f, wsum = 0.0f;
    for (int j = 0; j < Lwin; ++j) {
      int pos = idx + j - half;
      if (pos >= 0 && pos < C) {
        float w = win[j];
        dot  += w * base[pos];
        wsum += w;
      }
    }
    row_loss[row] = wsum * lse - dot;
  }
}

// Deterministic fixed-order mean over n row losses.
__global__ __launch_bounds__(NT) void xent_boeck_reduce(
    const float* __restrict__ row_loss, float* __restrict__ out, int n) {
  __shared__ float sm[NT];
  float a = 0.0f;
  for (int i = threadIdx.x; i < n; i += NT) a += row_loss[i];  // fixed per-thread order
  sm[threadIdx.x] = a;
  __syncthreads();
  for (int o = NT / 2; o > 0; o >>= 1) {
    if (threadIdx.x < o) sm[threadIdx.x] += sm[threadIdx.x + o];
    __syncthreads();
  }
  if (threadIdx.x == 0) out[0] = sm[0] / (float)n;
}

extern "C" void kernel_launch(void* const* d_in, const int* in_sizes, int n_in,
                              void* d_out, int out_size, void* d_ws, size_t ws_size,
                              hipStream_t stream) {
  const float* preds  = (const float*)d_in[0];
  const int*   labels = (const int*)d_in[1];
  const float* window = (const float*)d_in[2];
  const int B = in_sizes[1];
  const int C = in_sizes[0] / B;
  const int L = in_sizes[2];

  float* row_loss = (float*)d_ws;   // B floats of scratch
  float* out      = (float*)d_out;

  xent_boeck_row<<<B, NT, 0, stream>>>(preds, labels, window, L, row_loss, C);
  xent_boeck_reduce<<<1, NT, 0, stream>>>(row_loss, out, B);
}